// SSDIntNBitTableBatchedEmbeddingBags_3109556322443
// MI455X (gfx1250) — compile-verified
//
#include <hip/hip_runtime.h>
#include <stdint.h>

#ifndef __has_builtin
#define __has_builtin(x) 0
#endif

#if __has_builtin(__builtin_amdgcn_global_load_async_to_lds_b128)
#define HAVE_ASYNC_LDS 1
#else
#define HAVE_ASYNC_LDS 0
#endif

// Problem constants (from reference): T tables, ROWS x D uint8 rows,
// B bags of L lookups each.
constexpr int       kT     = 2;
constexpr int       kD     = 128;      // bytes per row (uint8 codes)
constexpr int       kB     = 4096;
constexpr int       kL     = 50;
constexpr long long kRows  = 1000000;
constexpr int       kGroups = (kL + 3) / 4;   // 13 groups of 4 rows
constexpr int       kWavesPerBlock = 8;       // 256 threads, wave32
constexpr int       kDepth = 4;               // async pipeline depth (ASYNCcnt <= 3)

// Types matching the async builtin's signature: (v4i AS1*, v4i AS3*, imm, imm)
typedef int v4i __attribute__((vector_size(16)));
typedef __attribute__((address_space(1))) v4i g1_v4i;
typedef __attribute__((address_space(3))) v4i l3_v4i;

static __device__ __forceinline__ void wait_async_le(int n) {
#if __has_builtin(__builtin_amdgcn_s_wait_asynccnt)
  switch (n) {
    case 0:  __builtin_amdgcn_s_wait_asynccnt(0); break;
    case 1:  __builtin_amdgcn_s_wait_asynccnt(1); break;
    case 2:  __builtin_amdgcn_s_wait_asynccnt(2); break;
    default: __builtin_amdgcn_s_wait_asynccnt(3); break;
  }
#else
  switch (n) {
    case 0:  asm volatile("s_wait_asynccnt 0" ::: "memory"); break;
    case 1:  asm volatile("s_wait_asynccnt 1" ::: "memory"); break;
    case 2:  asm volatile("s_wait_asynccnt 2" ::: "memory"); break;
    default: asm volatile("s_wait_asynccnt 3" ::: "memory"); break;
  }
#endif
  asm volatile("" ::: "memory");  // keep LDS reads below the wait
}

// Accumulate 16 dequantized bytes (one uint4) into acc[16] with scale s.
static __device__ __forceinline__ void accum16(float acc[16], uint4 q, float s) {
  const unsigned w[4] = {q.x, q.y, q.z, q.w};
#pragma unroll
  for (int j = 0; j < 4; ++j) {
#pragma unroll
    for (int k = 0; k < 4; ++k) {
      acc[j * 4 + k] += (float)((w[j] >> (8 * k)) & 0xffu) * s;  // v_cvt_f32_ubyteN + v_fma
    }
  }
}

// Shared epilogue: cross-quarter reduction + bias + 512B coalesced store.
static __device__ __forceinline__ void reduce_and_store(
    float acc[16], float bias_acc, float* __restrict__ out,
    int b, int t, int quarter, int sub) {
#pragma unroll
  for (int j = 0; j < 16; ++j) {
    acc[j] += __shfl_xor(acc[j], 8, 32);    // combine rows 0<->1, 2<->3
    acc[j] += __shfl_xor(acc[j], 16, 32);   // combine row pairs
  }
#pragma unroll
  for (int off = 16; off >= 1; off >>= 1)
    bias_acc += __shfl_xor(bias_acc, off, 32);
#pragma unroll
  for (int j = 0; j < 16; ++j) acc[j] += bias_acc;

  // Element d = sub*16 + accIdx; quarter q stores float4 #q of this chunk.
  float4 v = make_float4(acc[quarter * 4 + 0], acc[quarter * 4 + 1],
                         acc[quarter * 4 + 2], acc[quarter * 4 + 3]);
  float* op = out + (size_t)b * (kT * kD) + (size_t)t * kD + sub * 16 + quarter * 4;
  *(float4*)op = v;   // 32 lanes x 16B = full 512B row of output
}

#if HAVE_ASYNC_LDS
// ---------------- Async-LDS 4-deep pipelined kernel (CDNA5 path) ------------
// One wave per bag. Each "group" = 4 rows; lane (quarter,sub) covers 16B of
// one row. Rows are DMA'd global->LDS with global_load_async_to_lds_b128
// (no VGPR destination), retired in order via s_wait_asynccnt, then
// dequantized from LDS. 4 groups (2KB/wave) kept in flight to cover random
// HBM gather latency.
__global__ __launch_bounds__(256) void tbe_async(
    const uint8_t* __restrict__ qw, const float* __restrict__ sc,
    const float* __restrict__ bs, const int* __restrict__ ind,
    float* __restrict__ out) {
  __shared__ __align__(16) uint8_t stage[kWavesPerBlock][kDepth][4 * kD];  // 16KB

  const int lane    = threadIdx.x & 31;
  const int wib     = threadIdx.x >> 5;
  const int bag     = blockIdx.x * kWavesPerBlock + wib;  // 0 .. T*B-1
  const int t       = bag >> 12;                          // B == 4096
  const int b       = bag & (kB - 1);
  const int quarter = lane >> 3;   // which of 4 rows in the group
  const int sub     = lane & 7;    // which 16B chunk of the 128B row

  const int*     bag_idx = ind + (size_t)bag * kL;
  const uint8_t* table   = qw + (size_t)t * kRows * kD;
  const float*   s_tab   = sc + (size_t)t * kRows;
  const float*   b_tab   = bs + (size_t)t * kRows;

  float acc[16];
#pragma unroll
  for (int j = 0; j < 16; ++j) acc[j] = 0.f;

  float bias_acc = 0.f;
  for (int l = lane; l < kL; l += 32) bias_acc += b_tab[bag_idx[l]];

  int   rsave[kDepth];
  float ssave[kDepth];

  auto issue = [&](int g) {
    const int slot = g % kDepth;
    const int l = g * 4 + quarter;
    int   r = -1;
    float s = 0.f;
    if (l < kL) {
      r = bag_idx[l];
      s = s_tab[r];  // plain vmem load, latency hidden behind async window
      const uint8_t* src = table + (size_t)r * kD + sub * 16;
      uint8_t*       dst = &stage[wib][slot][quarter * kD + sub * 16];
      // global -> LDS async copy, 16B/lane, 512B per wave-op; ASYNCcnt += 1
      __builtin_amdgcn_global_load_async_to_lds_b128(
          (g1_v4i*)(uintptr_t)src,   // global address (generic == AS1 value)
          (l3_v4i*)dst,              // addrspacecast generic -> LDS offset
          0, 0);
    }
    rsave[slot] = r;
    ssave[slot] = s;
  };

  // Prologue: fill the pipeline (groups 0..kDepth-2).
#pragma unroll
  for (int p = 0; p < kDepth - 1; ++p) issue(p);

#pragma unroll
  for (int g = 0; g < kGroups; ++g) {
    if (g + kDepth - 1 < kGroups) issue(g + kDepth - 1);
    // Async loads retire in order: group g done when outstanding <= remaining.
    const int rem = kGroups - 1 - g;
    wait_async_le(rem < (kDepth - 1) ? rem : (kDepth - 1));
    const int slot = g % kDepth;
    if (rsave[slot] >= 0) {
      const uint4 q = *(const uint4*)&stage[wib][slot][quarter * kD + sub * 16];
      accum16(acc, q, ssave[slot]);
    }
  }
  reduce_and_store(acc, bias_acc, out, b, t, quarter, sub);
}

#else  // !HAVE_ASYNC_LDS ---------------------------------------------------
// Fallback: direct vector-load kernel (same layout, no LDS staging).
__global__ __launch_bounds__(256) void tbe_direct(
    const uint8_t* __restrict__ qw, const float* __restrict__ sc,
    const float* __restrict__ bs, const int* __restrict__ ind,
    float* __restrict__ out) {
  const int lane    = threadIdx.x & 31;
  const int wib     = threadIdx.x >> 5;
  const int bag     = blockIdx.x * kWavesPerBlock + wib;
  const int t       = bag >> 12;
  const int b       = bag & (kB - 1);
  const int quarter = lane >> 3;
  const int sub     = lane & 7;

  const int*     bag_idx = ind + (size_t)bag * kL;
  const uint8_t* table   = qw + (size_t)t * kRows * kD;
  const float*   s_tab   = sc + (size_t)t * kRows;
  const float*   b_tab   = bs + (size_t)t * kRows;

  float acc[16];
#pragma unroll
  for (int j = 0; j < 16; ++j) acc[j] = 0.f;

  float bias_acc = 0.f;
  for (int l = lane; l < kL; l += 32) bias_acc += b_tab[bag_idx[l]];

  for (int l0 = 0; l0 < kL; l0 += 4) {
    const int l = l0 + quarter;
    const int ln = l + 4;
    if (ln < kL) __builtin_prefetch(table + (size_t)bag_idx[ln] * kD + sub * 16, 0, 0);
    if (l < kL) {
      const int r = bag_idx[l];
      const uint4 q = *(const uint4*)(table + (size_t)r * kD + sub * 16);
      accum16(acc, q, s_tab[r]);
    }
  }
  reduce_and_store(acc, bias_acc, out, b, t, quarter, sub);
}
#endif  // HAVE_ASYNC_LDS

extern "C" void kernel_launch(void* const* d_in, const int* in_sizes, int n_in,
                              void* d_out, int out_size, void* d_ws, size_t ws_size,
                              hipStream_t stream) {
  const uint8_t* qw  = (const uint8_t*)d_in[0];  // [T, ROWS, D] uint8
  const float*   sc  = (const float*)d_in[1];    // [T, ROWS] f32
  const float*   bs  = (const float*)d_in[2];    // [T, ROWS] f32
  const int*     ind = (const int*)d_in[3];      // [T, B, L] i32
  float*         out = (float*)d_out;            // [B, T*D] f32

  const int bags = kT * kB;                      // 8192 waves, one per bag
  dim3 grid(bags / kWavesPerBlock), block(kWavesPerBlock * 32);
#if HAVE_ASYNC_LDS
  tbe_async<<<grid, block, 0, stream>>>(qw, sc, bs, ind, out);
#else
  tbe_direct<<<grid, block, 0, stream>>>(qw, sc, bs, ind, out);
#endif
}